// GNN_28071906247359
// MI455X (gfx1250) — compile-verified
//
#include <hip/hip_runtime.h>

// GNN: 4 layers of  m=relu(h[src]W^T+b); mm=segmax_dst(m); h=rmsnorm(h+mm);
//      y=relu(hW2^T+b2); h=rmsnorm(h+y)
// Key transform: relu(h[src]@W^T+b) == relu(h@W^T+b)[src]  -> compute per-node,
// scatter-max per-edge with u32 atomics on non-negative float bit patterns.
// Zero-init of mm implements the deg==0 -> 0 masking of the reference exactly.

#define N_NODES 40000
#define N_EDGES 640000
#define DIM     128
#define LAYERS  4
#define EPS_RMS 1e-6f

typedef __attribute__((ext_vector_type(2))) float v2f;
typedef __attribute__((ext_vector_type(8))) float v8f;

// ---------- W (o,d) -> K-pair-interleaved transpose ----------
// Element (k, o) of W^T stored at Wtp[(k>>1)*256 + 2*o + (k&1)] so the WMMA
// B-operand fetch {B[k][o], B[k+1][o]} is one contiguous 8-byte load per lane.
__global__ __launch_bounds__(256) void transpose_w(const float* __restrict__ W,
                                                   float* __restrict__ Wtp) {
    int idx = blockIdx.x * 256 + threadIdx.x;   // 0..16383, coalesced read
    int o = idx >> 7, d = idx & 127;
    Wtp[(d >> 1) * 256 + 2 * o + (d & 1)] = W[idx];   // W[idx] == W[o*128 + d]
}

// ---------- zero fill (graph-capture-safe memset) ----------
__global__ __launch_bounds__(256) void zero_f4(float4* __restrict__ p, int n4) {
    int i = blockIdx.x * 256 + threadIdx.x;
    if (i < n4) p[i] = make_float4(0.f, 0.f, 0.f, 0.f);
}

// ---------- out = relu(h @ W^T + bias), N x 128, f32 WMMA 16x16x4 ----------
// Block: 256 thr = 8 waves. Block owns 16 output rows; wave w owns cols [16w,16w+16).
// A tile (16x128 of h) staged in LDS with padded stride 132 (bank-conflict free).
__global__ __launch_bounds__(256)
void matmul_relu_wmma(const float* __restrict__ h,
                      const float* __restrict__ Wtp,  // K-pair interleaved W^T
                      const float* __restrict__ bias,
                      float* __restrict__ out) {
    __shared__ float hT[16 * 132];
    const int tid  = threadIdx.x;
    const int row0 = blockIdx.x * 16;

    for (int i = tid; i < 16 * DIM; i += 256) {
        int m = i >> 7, k = i & 127;
        hT[m * 132 + k] = h[(size_t)(row0 + m) * DIM + k];
    }
    __syncthreads();

    const int wave = tid >> 5;
    const int lane = tid & 31;
    const int col0 = wave * 16;
    const int half = lane >> 4;     // 0: lanes 0-15, 1: lanes 16-31
    const int l15  = lane & 15;

    // float2 view: pair p, column o lives at index p*128 + o
    const float2* Wp2 = (const float2*)Wtp;

    v8f c = {0.f, 0.f, 0.f, 0.f, 0.f, 0.f, 0.f, 0.f};

#pragma unroll
    for (int k0 = 0; k0 < DIM; k0 += 4) {
        const int kk = k0 + 2 * half;   // even
        // A 16x4 (MxK): lane m<16 -> {K=k0,k0+1}, lane m+16 -> {K=k0+2,k0+3}
        v2f a;
        a.x = hT[l15 * 132 + kk];
        a.y = hT[l15 * 132 + kk + 1];
        // B 4x16 (KxN): lanes 0-15 -> rows k0,k0+1 ; lanes 16-31 -> rows k0+2,k0+3
        const float2 bp = Wp2[(kk >> 1) * 128 + col0 + l15];  // one b64 load
        v2f b;
        b.x = bp.x;
        b.y = bp.y;
        // 8 args: (neg_a, A, neg_b, B, c_mod, C, reuse_a, reuse_b)
        c = __builtin_amdgcn_wmma_f32_16x16x4_f32(false, a, false, b,
                                                  (short)0, c, false, false);
    }

    const float bv = bias[col0 + l15];
#pragma unroll
    for (int v = 0; v < 8; ++v) {
        // C/D layout: VGPR v -> M = v (lanes 0-15) / v+8 (lanes 16-31), N = lane&15
        const int row = row0 + v + 8 * half;
        const int col = col0 + l15;
        out[(size_t)row * DIM + col] = fmaxf(c[v] + bv, 0.f);
    }
}

// ---------- per-edge scatter-max: mm[dst] = max(mm[dst], t[src]) ----------
// One wave per edge; lane handles 4 consecutive floats (uint4 bit patterns,
// valid since t = relu(...) >= 0).
__global__ __launch_bounds__(256)
void scatter_max_edges(const float* __restrict__ t,
                       const int2* __restrict__ e,
                       unsigned int* __restrict__ mm) {
    const int edge = blockIdx.x * 8 + (threadIdx.x >> 5);  // E % 8 == 0
    const int lane = threadIdx.x & 31;
    const int2 sd = e[edge];                               // wave-uniform b64 load
    const uint4 v = ((const uint4*)(t + (size_t)sd.x * DIM))[lane];
    unsigned int* p = mm + (size_t)sd.y * DIM + 4 * lane;
    atomicMax(p + 0, v.x);
    atomicMax(p + 1, v.y);
    atomicMax(p + 2, v.z);
    atomicMax(p + 3, v.w);
}

// ---------- out = rmsnorm(hin + add) * g + bg ; one wave32 per row ----------
__global__ __launch_bounds__(256)
void add_rmsnorm(const float* __restrict__ hin,
                 const float* __restrict__ add,
                 const float* __restrict__ g,
                 const float* __restrict__ bg,
                 float* __restrict__ out) {
    const int row  = blockIdx.x * 8 + (threadIdx.x >> 5); // N % 8 == 0
    const int lane = threadIdx.x & 31;
    const float4 a = ((const float4*)(hin + (size_t)row * DIM))[lane];
    const float4 b = ((const float4*)(add + (size_t)row * DIM))[lane];
    float4 s = {a.x + b.x, a.y + b.y, a.z + b.z, a.w + b.w};
    float ss = s.x * s.x + s.y * s.y + s.z * s.z + s.w * s.w;
#pragma unroll
    for (int off = 16; off > 0; off >>= 1)
        ss += __shfl_xor(ss, off, 32);                    // wave32 reduction
    const float r = rsqrtf(ss * (1.0f / DIM) + EPS_RMS);
    const int col = 4 * lane;
    float4 o;
    o.x = s.x * r * g[col + 0] + bg[col + 0];
    o.y = s.y * r * g[col + 1] + bg[col + 1];
    o.z = s.z * r * g[col + 2] + bg[col + 2];
    o.w = s.w * r * g[col + 3] + bg[col + 3];
    ((float4*)(out + (size_t)row * DIM))[lane] = o;
}

extern "C" void kernel_launch(void* const* d_in, const int* in_sizes, int n_in,
                              void* d_out, int out_size, void* d_ws, size_t ws_size,
                              hipStream_t stream) {
    const float* x   = (const float*)d_in[0];
    const int2*  e   = (const int2*) d_in[1];
    const float* Wc  = (const float*)d_in[2];
    const float* bc  = (const float*)d_in[3];
    const float* gc  = (const float*)d_in[4];
    const float* bgc = (const float*)d_in[5];
    const float* Wh  = (const float*)d_in[6];
    const float* bh  = (const float*)d_in[7];
    const float* gh  = (const float*)d_in[8];
    const float* bgh = (const float*)d_in[9];

    float* h   = (float*)d_out;                      // persistent node state
    float* t   = (float*)d_ws;                       // relu(matmul) output
    float* mm  = t  + (size_t)N_NODES * DIM;         // scatter-max accumulator / hmid
    float* Wtp = mm + (size_t)N_NODES * DIM;         // 128x128 pair-interleaved W^T

    const int mm_blocks = (N_NODES * DIM / 4) / 256; // 5000
    const int mm_grid   = N_NODES / 16;              // 2500
    const int row_grid  = N_NODES / 8;               // 5000
    const int edge_grid = N_EDGES / 8;               // 80000

    for (int l = 0; l < LAYERS; ++l) {
        const float* hin = (l == 0) ? x : h;

        // m_node = relu(h @ Wc^T + bc)
        transpose_w<<<64, 256, 0, stream>>>(Wc + (size_t)l * DIM * DIM, Wtp);
        matmul_relu_wmma<<<mm_grid, 256, 0, stream>>>(hin, Wtp, bc + l * DIM, t);

        // mm = segment_max over edges (zero-init == deg==0 masking)
        zero_f4<<<mm_blocks, 256, 0, stream>>>((float4*)mm, N_NODES * DIM / 4);
        scatter_max_edges<<<edge_grid, 256, 0, stream>>>(t, e, (unsigned int*)mm);

        // hmid = rmsnorm(h + mm) * gc + bgc   (in-place into mm; wave-private rows)
        add_rmsnorm<<<row_grid, 256, 0, stream>>>(hin, mm, gc + l * DIM, bgc + l * DIM, mm);

        // y = relu(hmid @ Wh^T + bh);  h = rmsnorm(hmid + y) * gh + bgh
        transpose_w<<<64, 256, 0, stream>>>(Wh + (size_t)l * DIM * DIM, Wtp);
        matmul_relu_wmma<<<mm_grid, 256, 0, stream>>>(mm, Wtp, bh + l * DIM, t);
        add_rmsnorm<<<row_grid, 256, 0, stream>>>(mm, t, gh + l * DIM, bgh + l * DIM, h);
    }
}